// Generator_79001628442695
// MI455X (gfx1250) — compile-verified
//
#include <hip/hip_runtime.h>
#include <math.h>

// ---------------------------------------------------------------------------
// CDNA5 (gfx1250) HoloGAN generator pipeline, v2.
//  * All GEMM-shaped work on V_WMMA_F32_16X16X4_F32 (fp32, M = batch = 16).
//  * Activations kept channels-last so every WMMA A-operand is ONE aligned
//    b64 load; weights pre-packed so every B-operand is ONE coalesced b64.
//  * global_prefetch (via __builtin_prefetch) pulls the next weight tap.
// ---------------------------------------------------------------------------

typedef __attribute__((ext_vector_type(2))) float v2f;
typedef __attribute__((ext_vector_type(8))) float v8f;

__device__ __forceinline__ float wave_sum32(float v) {
#pragma unroll
  for (int off = 16; off > 0; off >>= 1) v += __shfl_xor(v, off, 32);
  return v;
}

__device__ __forceinline__ v8f wmma_f32x4(v2f a, v2f b, v8f c) {
  return __builtin_amdgcn_wmma_f32_16x16x4_f32(false, a, false, b, (short)0, c,
                                               false, false);
}

// ---------------------------------------------------------------------------
// Style MLP:  out = relu(z @ W^T + b),  z:(16,128)  W:(N,128)  out:(16,N)
// ---------------------------------------------------------------------------
__global__ void zmap_wmma_kernel(const float* __restrict__ z,
                                 const float* __restrict__ W,
                                 const float* __restrict__ bias,
                                 float* __restrict__ out, int N) {
  const int lane = threadIdx.x;
  const int half = lane >> 4;
  const int mn   = lane & 15;
  const int kb   = half * 2;
  const int n0   = blockIdx.x * 16;

  v8f acc = {};
#pragma unroll 4
  for (int k0 = 0; k0 < 128; k0 += 4) {
    const v2f a = *(const v2f*)(z + mn * 128 + k0 + kb);
    const v2f b = *(const v2f*)(W + (n0 + mn) * 128 + k0 + kb);
    acc = wmma_f32x4(a, b, acc);
  }
#pragma unroll
  for (int r = 0; r < 8; ++r) {
    const int m = r + half * 8;
    const int n = n0 + mn;
    const float v = acc[r] + bias[n];
    out[m * N + n] = v > 0.f ? v : 0.f;
  }
}

// ---------------------------------------------------------------------------
// Weight repack:  w(Cin, Cout, k2)  ->  wp[((tap*(Cin/2)+cp)*Cout+co)*2+p]
// so a WMMA B row-pair is one aligned, lane-coalesced b64 load.
// ---------------------------------------------------------------------------
__global__ void pack_w_kernel(const float* __restrict__ w,
                              float* __restrict__ wp, int Cin, int Cout,
                              int k2) {
  const int idx = blockIdx.x * 256 + threadIdx.x;
  const int total = Cin * Cout * k2;
  if (idx >= total) return;
  int t = idx;
  const int p = t & 1;          t >>= 1;
  const int co = t % Cout;      t /= Cout;
  const int cp = t % (Cin / 2); t /= (Cin / 2);
  const int tap = t;
  wp[idx] = w[((2 * cp + p) * Cout + co) * k2 + tap];
}

// ---------------------------------------------------------------------------
// AdaIN(+ReLU) on channels-last activations, in place. One wave per (b,c).
// buf layout: [b][sp(S)][c(C)]
// ---------------------------------------------------------------------------
__global__ void adain_relu_cl_kernel(float* __restrict__ buf,
                                     const float* __restrict__ st, int C,
                                     int S) {
  const int c = blockIdx.x % C;
  const int b = blockIdx.x / C;
  float* p = buf + b * S * C + c;
  const int lane = threadIdx.x;

  float sum = 0.f, sq = 0.f;
  for (int i = lane; i < S; i += 32) {
    const float v = p[i * C];
    sum += v;
    sq += v * v;
  }
  sum = wave_sum32(sum);
  sq  = wave_sum32(sq);
  const float mean = sum / (float)S;
  const float var  = (sq - sum * mean) / (float)(S - 1);
  const float rs   = rsqrtf(var + 1e-8f);
  const float sc   = st[b * 2 * C + c];
  const float bi   = st[b * 2 * C + C + c];
  for (int i = lane; i < S; i += 32) {
    const float v = sc * (p[i * C] - mean) * rs + bi;
    p[i * C] = v > 0.f ? v : 0.f;
  }
}

// h0 stage: src = const_w (C,S) NCHW, broadcast over batch; dst channels-last.
__global__ void adain_bcast_kernel(const float* __restrict__ src,
                                   float* __restrict__ dst,
                                   const float* __restrict__ st, int C,
                                   int S) {
  const int c = blockIdx.x % C;
  const int b = blockIdx.x / C;
  const float* p = src + c * S;
  const int lane = threadIdx.x;

  float sum = 0.f, sq = 0.f;
  for (int i = lane; i < S; i += 32) {
    const float v = p[i];
    sum += v;
    sq += v * v;
  }
  sum = wave_sum32(sum);
  sq  = wave_sum32(sq);
  const float mean = sum / (float)S;
  const float var  = (sq - sum * mean) / (float)(S - 1);
  const float rs   = rsqrtf(var + 1e-8f);
  const float sc   = st[b * 2 * C + c];
  const float bi   = st[b * 2 * C + C + c];
  for (int i = lane; i < S; i += 32) {
    const float v = sc * (p[i] - mean) * rs + bi;
    dst[(b * S + i) * C + c] = v > 0.f ? v : 0.f;
  }
}

// ---------------------------------------------------------------------------
// 3-D transposed conv (k=3, s=2, p=1) as implicit GEMM on channels-last data.
// x: [b][sp_in][Cin] CL, wp packed, out: [b][sp_out][Cout] CL.
// ---------------------------------------------------------------------------
__global__ void convt3d_wmma_kernel(const float* __restrict__ x,
                                    const float* __restrict__ wp,
                                    const float* __restrict__ bias,
                                    float* __restrict__ out, int Cin, int Cout,
                                    int inS, int outS) {
  const int lane = threadIdx.x, half = lane >> 4, mn = lane & 15;
  const int sp = blockIdx.x;
  const int outS2 = outS * outS;
  const int ow = sp % outS, oh = (sp / outS) % outS, od = sp / outS2;
  const int n0 = blockIdx.y * 16;
  const int inS2 = inS * inS, inS3 = inS2 * inS;
  const int Cin2 = Cin >> 1;

  int tda[2], ida[2], nda = 0;
  int tha[2], iha[2], nha = 0;
  int twa[2], iwa[2], nwa = 0;
  for (int t = 0; t < 3; ++t) {
    int num = od + 1 - t;
    if (num >= 0 && (num & 1) == 0 && (num >> 1) < inS) { tda[nda] = t; ida[nda] = num >> 1; ++nda; }
    num = oh + 1 - t;
    if (num >= 0 && (num & 1) == 0 && (num >> 1) < inS) { tha[nha] = t; iha[nha] = num >> 1; ++nha; }
    num = ow + 1 - t;
    if (num >= 0 && (num & 1) == 0 && (num >> 1) < inS) { twa[nwa] = t; iwa[nwa] = num >> 1; ++nwa; }
  }

  const int co = n0 + mn;
  v8f acc = {};
  for (int a = 0; a < nda; ++a)
    for (int b2 = 0; b2 < nha; ++b2)
      for (int c2 = 0; c2 < nwa; ++c2) {
        const int tap = (tda[a] * 3 + tha[b2]) * 3 + twa[c2];
        const int isp = (ida[a] * inS + iha[b2]) * inS + iwa[c2];
        const float* xa = x + (mn * inS3 + isp) * Cin + half * 2;   // A base
        const float* wb = wp + ((tap * Cin2 + half) * Cout + co) * 2;  // B base
        __builtin_prefetch(wp + (((tap + 1) * Cin2) * Cout + co) * 2, 0, 1);
        for (int ci = 0; ci < Cin; ci += 4) {
          const v2f av = *(const v2f*)(xa + ci);
          const v2f bv = *(const v2f*)(wb + ci * Cout);  // (ci/2)*Cout*2
          acc = wmma_f32x4(av, bv, acc);
        }
      }

  const int outS3 = outS2 * outS;
  const float bco = bias[co];
#pragma unroll
  for (int r = 0; r < 8; ++r) {
    const int m = r + half * 8;
    out[(m * outS3 + sp) * Cout + co] = acc[r] + bco;
  }
}

// ---------------------------------------------------------------------------
// Generic 2-D transposed conv (1x1 projection, 4x4/s2 blocks, 4x4/s1 output).
// x channels-last; wp packed.  mode: 0=linear 1=relu 2=tanh.
// out_nchw: final layer writes NCHW into d_out, else channels-last.
// ---------------------------------------------------------------------------
__global__ void convt2d_wmma_kernel(const float* __restrict__ x,
                                    const float* __restrict__ wp,
                                    const float* __restrict__ bias,
                                    float* __restrict__ out, int Cin, int Cout,
                                    int inS, int outS, int k, int stride,
                                    int pad, int mode, int out_nchw) {
  const int lane = threadIdx.x, half = lane >> 4, mn = lane & 15;
  const int sp = blockIdx.x;
  const int ow = sp % outS, oh = sp / outS;
  const int n0 = blockIdx.y * 16;
  const int Cin2 = Cin >> 1;

  int tha[4], iha[4], nha = 0;
  int twa[4], iwa[4], nwa = 0;
  for (int t = 0; t < k; ++t) {
    int num = oh + pad - t;
    if (num >= 0 && num % stride == 0 && (num / stride) < inS) { tha[nha] = t; iha[nha] = num / stride; ++nha; }
    num = ow + pad - t;
    if (num >= 0 && num % stride == 0 && (num / stride) < inS) { twa[nwa] = t; iwa[nwa] = num / stride; ++nwa; }
  }

  const int co = n0 + mn;
  const int co_ld = co < Cout ? co : (Cout - 1);
  v8f acc = {};
  for (int a = 0; a < nha; ++a)
    for (int b2 = 0; b2 < nwa; ++b2) {
      const int tap = tha[a] * k + twa[b2];
      const int isp = iha[a] * inS + iwa[b2];
      const float* xa = x + (mn * inS * inS + isp) * Cin + half * 2;
      const float* wb = wp + ((tap * Cin2 + half) * Cout + co_ld) * 2;
      __builtin_prefetch(wp + (((tap + 1) * Cin2) * Cout + co_ld) * 2, 0, 1);
      for (int ci = 0; ci < Cin; ci += 4) {
        const v2f av = *(const v2f*)(xa + ci);
        const v2f bv = *(const v2f*)(wb + ci * Cout);
        acc = wmma_f32x4(av, bv, acc);
      }
    }

  if (co < Cout) {
    const int o2 = outS * outS;
    const float bco = bias[co];
#pragma unroll
    for (int r = 0; r < 8; ++r) {
      const int m = r + half * 8;
      float v = acc[r] + bco;
      if (mode == 1) v = v > 0.f ? v : 0.f;
      else if (mode == 2) v = tanhf(v);
      if (out_nchw) out[(m * Cout + co) * o2 + sp] = v;
      else          out[(m * o2 + sp) * Cout + co] = v;
    }
  }
}

// ---------------------------------------------------------------------------
// Per-batch inverse affine view matrix (adjugate of the 3x3 linear part).
// ---------------------------------------------------------------------------
__global__ void transform_kernel(const float* __restrict__ view,
                                 float* __restrict__ Amat) {
  const int b = threadIdx.x;
  if (b >= 16) return;
  const float th = view[b * 6 + 0], ga = view[b * 6 + 1], sc = view[b * 6 + 2];
  const float xs = view[b * 6 + 3], ys = view[b * 6 + 4], zs = view[b * 6 + 5];
  const float c = cosf(th), s = sinf(th);
  const float cg = cosf(ga), sg = sinf(ga);
  const float R00 = cg * c,  R01 = sg,  R02 = -cg * s;
  const float R10 = -sg * c, R11 = cg,  R12 = sg * s;
  const float R20 = s,       R21 = 0.f, R22 = c;
  const float L00 = sc * R00, L01 = sc * R01, L02 = sc * R02;
  const float L10 = sc * R10, L11 = sc * R11, L12 = sc * R12;
  const float L20 = sc * R20, L21 = sc * R21, L22 = sc * R22;
  const float hs = 8.f;
  const float ft0 = hs + xs - hs * (L00 + L01 + L02);
  const float ft1 = hs + ys - hs * (L10 + L11 + L12);
  const float ft2 = hs + zs - hs * (L20 + L21 + L22);
  const float det = L00 * (L11 * L22 - L12 * L21) -
                    L01 * (L10 * L22 - L12 * L20) +
                    L02 * (L10 * L21 - L11 * L20);
  const float id_ = 1.f / det;
  const float I00 =  (L11 * L22 - L12 * L21) * id_;
  const float I01 = -(L01 * L22 - L02 * L21) * id_;
  const float I02 =  (L01 * L12 - L02 * L11) * id_;
  const float I10 = -(L10 * L22 - L12 * L20) * id_;
  const float I11 =  (L00 * L22 - L02 * L20) * id_;
  const float I12 = -(L00 * L12 - L02 * L10) * id_;
  const float I20 =  (L10 * L21 - L11 * L20) * id_;
  const float I21 = -(L00 * L21 - L01 * L20) * id_;
  const float I22 =  (L00 * L11 - L01 * L10) * id_;
  float* A = Amat + b * 12;
  A[0] = I00; A[1] = I01; A[2]  = I02; A[3]  = -(I00 * ft0 + I01 * ft1 + I02 * ft2);
  A[4] = I10; A[5] = I11; A[6]  = I12; A[7]  = -(I10 * ft0 + I11 * ft1 + I12 * ft2);
  A[8] = I20; A[9] = I21; A[10] = I22; A[11] = -(I20 * ft0 + I21 * ft1 + I22 * ft2);
}

// ---------------------------------------------------------------------------
// Trilinear resample reproducing the reference's exact reshape semantics.
// Reference gathers vf[row, c] = h2_NCDHW_flat[row*64 + c]; our h2 is
// channels-last, so translate: g = row_in_batch*64+c -> cc=g>>12, sp=g&4095.
// Output feeds the 1x1 conv as (b, 1024ch, 256sp); we store it channels-last
// for that conv: f = n*64+c -> ch=f>>8, s=f&255 -> dst[(b*256+s)*1024+ch].
// ---------------------------------------------------------------------------
__global__ void sample3d_kernel(const float* __restrict__ voxcl,
                                const float* __restrict__ Amat,
                                float* __restrict__ dst) {
  const int blk = blockIdx.x;
  const int b = blk >> 12;
  const int n = blk & 4095;
  const int xx = n & 15, yy = (n >> 4) & 15, zz = n >> 8;
  const float* A = Amat + b * 12;
  const float fx = A[0] * xx + A[1] * yy + A[2] * zz + A[3];
  const float fy = A[4] * xx + A[5] * yy + A[6] * zz + A[7];
  const float fz = A[8] * xx + A[9] * yy + A[10] * zz + A[11];
  int x0 = (int)floorf(fx), y0 = (int)floorf(fy), z0 = (int)floorf(fz);
  int x1 = x0 + 1, y1 = y0 + 1, z1 = z0 + 1;
  x0 = min(max(x0, 0), 15); x1 = min(max(x1, 0), 15);
  y0 = min(max(y0, 0), 15); y1 = min(max(y1, 0), 15);
  z0 = min(max(z0, 0), 15); z1 = min(max(z1, 0), 15);
  const float x0f = (float)x0, x1f = (float)x1;
  const float y0f = (float)y0, y1f = (float)y1;
  const float z0f = (float)z0, z1f = (float)z1;
  const float wa = (x1f - fx) * (y1f - fy) * (z1f - fz);
  const float wb = (x1f - fx) * (fy - y0f) * (z1f - fz);
  const float wc = (fx - x0f) * (y1f - fy) * (z1f - fz);
  const float wd = (fx - x0f) * (fy - y0f) * (z1f - fz);
  const float we = (x1f - fx) * (y1f - fy) * (fz - z0f);
  const float wf = (x1f - fx) * (fy - y0f) * (fz - z0f);
  const float wg = (fx - x0f) * (y1f - fy) * (fz - z0f);
  const float wh = (fx - x0f) * (fy - y0f) * (fz - z0f);
  const int c = threadIdx.x;
  const float* vb = voxcl + b * 4096 * 64;
#define G(zi, yi, xi)                                              \
  ({                                                               \
    const int g = (((zi)*256 + (yi)*16 + (xi)) << 6) + c;          \
    vb[(g & 4095) * 64 + (g >> 12)];                               \
  })
  const float v = wa * G(z0, y0, x0) + wb * G(z0, y1, x0) +
                  wc * G(z0, y0, x1) + wd * G(z0, y1, x1) +
                  we * G(z1, y0, x0) + wf * G(z1, y1, x0) +
                  wg * G(z1, y0, x1) + wh * G(z1, y1, x1);
#undef G
  const int f = (n << 6) + c;          // tgt-flat index within batch
  dst[(b * 256 + (f & 255)) * 1024 + (f >> 8)] = v;
}

// ---------------------------------------------------------------------------
extern "C" void kernel_launch(void* const* d_in, const int* in_sizes, int n_in,
                              void* d_out, int out_size, void* d_ws,
                              size_t ws_size, hipStream_t stream) {
  (void)in_sizes; (void)n_in; (void)out_size; (void)ws_size;
  const float* z      = (const float*)d_in[0];
  const float* view   = (const float*)d_in[1];
  const float* cw     = (const float*)d_in[2];
  const float* zm0_w  = (const float*)d_in[3];
  const float* zm0_b  = (const float*)d_in[4];
  const float* b1_w   = (const float*)d_in[5];
  const float* b1_b   = (const float*)d_in[6];
  const float* b1_zw  = (const float*)d_in[7];
  const float* b1_zb  = (const float*)d_in[8];
  const float* b2_w   = (const float*)d_in[9];
  const float* b2_b   = (const float*)d_in[10];
  const float* b2_zw  = (const float*)d_in[11];
  const float* b2_zb  = (const float*)d_in[12];
  const float* p_w    = (const float*)d_in[13];
  const float* p_b    = (const float*)d_in[14];
  const float* b3_w   = (const float*)d_in[15];
  const float* b3_b   = (const float*)d_in[16];
  const float* b3_zw  = (const float*)d_in[17];
  const float* b3_zb  = (const float*)d_in[18];
  const float* b4_w   = (const float*)d_in[19];
  const float* b4_b   = (const float*)d_in[20];
  const float* b4_zw  = (const float*)d_in[21];
  const float* b4_zb  = (const float*)d_in[22];
  const float* out_w  = (const float*)d_in[23];
  const float* out_b  = (const float*)d_in[24];

  float* ws = (float*)d_ws;
  float* st0  = ws;                       // 16*1024
  float* st1  = st0 + 16 * 1024;          // 16*256
  float* st2  = st1 + 16 * 256;           // 16*128
  float* st3  = st2 + 16 * 128;           // 16*512
  float* st4  = st3 + 16 * 512;           // 16*128
  float* amat = st4 + 16 * 128;           // 192, padded to 256
  float* h0   = amat + 256;               // 16*64*512     (CL)
  float* h1   = h0 + 16 * 64 * 512;       // 16*512*128    (CL)
  float* h2   = h1 + 16 * 512 * 128;      // 16*4096*64    (CL)
  float* h2d  = h2 + 16 * 4096 * 64;      // 16*256*1024   (CL)
  float* h3   = h2d + 16 * 256 * 1024;    // 16*256*1024   (CL)
  float* h4   = h3 + 16 * 256 * 1024;     // 16*1024*256   (CL)
  float* h5   = h4 + 16 * 1024 * 256;     // 16*4096*64    (CL)
  float* wp1  = h5 + 16 * 4096 * 64;      // 512*128*27
  float* wp2  = wp1 + 512 * 128 * 27;     // 128*64*27
  float* wpp  = wp2 + 128 * 64 * 27;      // 1024*1024
  float* wp3  = wpp + 1024 * 1024;        // 1024*256*16
  float* wp4  = wp3 + 1024 * 256 * 16;    // 256*64*16
  float* wpo  = wp4 + 256 * 64 * 16;      // 64*3*16

  // ---- weight repacks (independent) ----
  pack_w_kernel<<<(512 * 128 * 27 + 255) / 256, 256, 0, stream>>>(b1_w, wp1, 512, 128, 27);
  pack_w_kernel<<<(128 * 64 * 27 + 255) / 256, 256, 0, stream>>>(b2_w, wp2, 128, 64, 27);
  pack_w_kernel<<<(1024 * 1024 + 255) / 256, 256, 0, stream>>>(p_w, wpp, 1024, 1024, 1);
  pack_w_kernel<<<(1024 * 256 * 16 + 255) / 256, 256, 0, stream>>>(b3_w, wp3, 1024, 256, 16);
  pack_w_kernel<<<(256 * 64 * 16 + 255) / 256, 256, 0, stream>>>(b4_w, wp4, 256, 64, 16);
  pack_w_kernel<<<(64 * 3 * 16 + 255) / 256, 256, 0, stream>>>(out_w, wpo, 64, 3, 16);

  // ---- style MLPs (WMMA f32) ----
  zmap_wmma_kernel<<<64, 32, 0, stream>>>(z, zm0_w, zm0_b, st0, 1024);
  zmap_wmma_kernel<<<16, 32, 0, stream>>>(z, b1_zw, b1_zb, st1, 256);
  zmap_wmma_kernel<<<8,  32, 0, stream>>>(z, b2_zw, b2_zb, st2, 128);
  zmap_wmma_kernel<<<32, 32, 0, stream>>>(z, b3_zw, b3_zb, st3, 512);
  zmap_wmma_kernel<<<8,  32, 0, stream>>>(z, b4_zw, b4_zb, st4, 128);

  // ---- view matrices ----
  transform_kernel<<<1, 32, 0, stream>>>(view, amat);

  // ---- h0 = relu(adain(broadcast(const_w))) -> CL ----
  adain_bcast_kernel<<<16 * 512, 32, 0, stream>>>(cw, h0, st0, 512, 64);

  // ---- h1 = relu(adain(convT3d(h0))) : 512->128, 4^3 -> 8^3 ----
  convt3d_wmma_kernel<<<dim3(512, 8), 32, 0, stream>>>(h0, wp1, b1_b, h1, 512, 128, 4, 8);
  adain_relu_cl_kernel<<<16 * 128, 32, 0, stream>>>(h1, st1, 128, 512);

  // ---- h2 = relu(adain(convT3d(h1))) : 128->64, 8^3 -> 16^3 ----
  convt3d_wmma_kernel<<<dim3(4096, 4), 32, 0, stream>>>(h1, wp2, b2_b, h2, 128, 64, 8, 16);
  adain_relu_cl_kernel<<<16 * 64, 32, 0, stream>>>(h2, st2, 64, 4096);

  // ---- 3-D view transform / trilinear resample ----
  sample3d_kernel<<<16 * 4096, 64, 0, stream>>>(h2, amat, h2d);

  // ---- h3 = relu(conv1x1(h2d)) : 1024->1024 over 16x16 ----
  convt2d_wmma_kernel<<<dim3(256, 64), 32, 0, stream>>>(
      h2d, wpp, p_b, h3, 1024, 1024, 16, 16, 1, 1, 0, /*relu*/ 1, /*CL*/ 0);

  // ---- h4 = relu(adain(convT2d(h3))) : 1024->256, 16 -> 32 ----
  convt2d_wmma_kernel<<<dim3(1024, 16), 32, 0, stream>>>(
      h3, wp3, b3_b, h4, 1024, 256, 16, 32, 4, 2, 1, /*linear*/ 0, /*CL*/ 0);
  adain_relu_cl_kernel<<<16 * 256, 32, 0, stream>>>(h4, st3, 256, 1024);

  // ---- h5 = relu(adain(convT2d(h4))) : 256->64, 32 -> 64 ----
  convt2d_wmma_kernel<<<dim3(4096, 4), 32, 0, stream>>>(
      h4, wp4, b4_b, h5, 256, 64, 32, 64, 4, 2, 1, /*linear*/ 0, /*CL*/ 0);
  adain_relu_cl_kernel<<<16 * 64, 32, 0, stream>>>(h5, st4, 64, 4096);

  // ---- out = tanh(convT2d(h5)) : 64->3, 64 -> 65, NCHW output ----
  convt2d_wmma_kernel<<<dim3(65 * 65, 1), 32, 0, stream>>>(
      h5, wpo, out_b, (float*)d_out, 64, 3, 64, 65, 4, 1, 1, /*tanh*/ 2, /*NCHW*/ 1);
}